// MSS_SS2D_57878979281523
// MI455X (gfx1250) — compile-verified
//
#include <hip/hip_runtime.h>
#include <math.h>

#define HW 2304
#define WD 48
#define NB 8

typedef float v2f __attribute__((ext_vector_type(2)));
typedef float v8f __attribute__((ext_vector_type(8)));

static inline int cdiv(int a, int b) { return (a + b - 1) / b; }

__device__ __forceinline__ int dir_idx(int dir, int l) {
  switch (dir) {
    case 1:  return (l % WD) * WD + (l / WD);
    case 2:  return (HW - 1) - l;
    case 3:  { int j = (HW - 1) - l; return (j % WD) * WD + (j / WD); }
    default: return l;
  }
}

__device__ __forceinline__ float sigf(float x) { return 1.f / (1.f + __expf(-x)); }

// ======================= WMMA f32 GEMM over pixels =======================
// out[b, outCoff+n, hw] = act( sum_k in[b, inCoff+k, dir(hw)] * w[n*Cin+k] + bias[n] )
// M = NB*HW = 18432. One wave per block; each wave computes TWO 16x16 tiles
// (32 pixels x 16 outputs). Weight tile staged in LDS (zero-padded past Cout)
// so the K loop is branch-free: 4 coalesced A loads + 2 LDS reads + 2 WMMAs.
__global__ void gemm_wmma_k(const float* __restrict__ in, const float* __restrict__ w,
                            const float* __restrict__ bias, float* __restrict__ out,
                            int Cin, int Cout, int inBstr, int inCoff,
                            int outBstr, int outCoff, int act, int dir)
{
  __shared__ float wt_s[16 * 193];   // 16 rows x (Cin<=192)+1 padded stride
  int lane = threadIdx.x;
  int half = lane >> 4;              // 0: K=k+0..1 , 1: K=k+2..3
  int r    = lane & 15;
  int Cp1  = Cin + 1;

  // stage B tile: rows n_local=0..15 of W, zero pad beyond Cout
  for (int idx = lane; idx < 16 * Cin; idx += 32) {
    int nl = idx / Cin;
    int kk = idx - nl * Cin;
    int ng = blockIdx.y * 16 + nl;
    wt_s[nl * Cp1 + kk] = (ng < Cout) ? w[(size_t)ng * Cin + kk] : 0.f;
  }
  __syncthreads();

  int m0  = blockIdx.x * 32 + r;     // M tile 0 row for this lane
  int m1  = m0 + 16;                 // M tile 1 row
  int b0  = m0 / HW, hw0 = m0 % HW;
  int b1  = m1 / HW, hw1 = m1 % HW;
  const float* ap0 = in + (size_t)b0 * inBstr + (size_t)inCoff * HW + dir_idx(dir, hw0);
  const float* ap1 = in + (size_t)b1 * inBstr + (size_t)inCoff * HW + dir_idx(dir, hw1);
  const float* wr  = wt_s + r * Cp1;

  v8f acc0 = {0.f, 0.f, 0.f, 0.f, 0.f, 0.f, 0.f, 0.f};
  v8f acc1 = {0.f, 0.f, 0.f, 0.f, 0.f, 0.f, 0.f, 0.f};
  for (int k = 0; k < Cin; k += 4) {
    int ka = k + 2 * half;
    v2f a0, a1, bv;
    a0.x = ap0[(size_t)ka * HW];
    a0.y = ap0[(size_t)(ka + 1) * HW];
    a1.x = ap1[(size_t)ka * HW];
    a1.y = ap1[(size_t)(ka + 1) * HW];
    bv.x = wr[ka];
    bv.y = wr[ka + 1];
    __builtin_prefetch(ap0 + (size_t)(ka + 8) * HW, 0, 0); // speculative A prefetch
    acc0 = __builtin_amdgcn_wmma_f32_16x16x4_f32(false, a0, false, bv, (short)0, acc0,
                                                 false, false);
    acc1 = __builtin_amdgcn_wmma_f32_16x16x4_f32(false, a1, false, bv, (short)0, acc1,
                                                 false, false);
  }

  int n = blockIdx.y * 16 + r;
  if (n < Cout) {
    float bb = bias ? bias[n] : 0.f;
#pragma unroll
    for (int j = 0; j < 8; ++j) {
      // C/D layout: VGPR j holds M = j (lanes 0-15) / M = j+8 (lanes 16-31)
      int row0 = blockIdx.x * 32 + j + 8 * half;
      int ob0  = row0 / HW, ohw0 = row0 % HW;
      float v0 = acc0[j] + bb;
      if (act == 1)      v0 = fmaxf(v0, 0.f);
      else if (act == 2) v0 = sigf(v0);
      out[(size_t)ob0 * outBstr + (size_t)(outCoff + n) * HW + ohw0] = v0;

      int row1 = row0 + 16;
      int ob1  = row1 / HW, ohw1 = row1 % HW;
      float v1 = acc1[j] + bb;
      if (act == 1)      v1 = fmaxf(v1, 0.f);
      else if (act == 2) v1 = sigf(v1);
      out[(size_t)ob1 * outBstr + (size_t)(outCoff + n) * HW + ohw1] = v1;
    }
  }
}

// ======================= depthwise conv (generic) =======================
__global__ void dwconv_k(const float* __restrict__ in, const float* __restrict__ wt,
                         const float* __restrict__ bias, float* __restrict__ out,
                         int Cch, int inBstr, int inCoff, int outBstr, int outCoff,
                         int ks, int pad, int dil, int act, int total)
{
  int idx = blockIdx.x * blockDim.x + threadIdx.x;
  if (idx >= total) return;
  int p = idx % HW;
  int c = (idx / HW) % Cch;
  int b = idx / (HW * Cch);
  int h = p / WD, ww = p % WD;
  const float* ip = in + (size_t)b * inBstr + (size_t)(inCoff + c) * HW;
  float s = bias ? bias[c] : 0.f;
  for (int kh = 0; kh < ks; ++kh) {
    int ih = h - pad + kh * dil;
    if (ih < 0 || ih >= WD) continue;
    for (int kw = 0; kw < ks; ++kw) {
      int iw = ww - pad + kw * dil;
      if (iw < 0 || iw >= WD) continue;
      s += ip[ih * WD + iw] * wt[(size_t)c * ks * ks + kh * ks + kw];
    }
  }
  if (act == 1) s = s * sigf(s); // SiLU
  out[(size_t)b * outBstr + (size_t)(outCoff + c) * HW + p] = s;
}

// ======================= frequency branch: truncated 6x6 rDFT low-pass ===
__global__ void freq_low_k(const float* __restrict__ x, float* __restrict__ low)
{
  __shared__ float xpl[HW];
  __shared__ float Gr[6 * WD], Gi[6 * WD];
  __shared__ float Yr[36], Yi[36];
  __shared__ float Qr[6 * WD], Qi[6 * WD];
  int bc  = blockIdx.x;
  int tid = threadIdx.x;
  const float* xp = x + (size_t)bc * HW;
  for (int i = tid; i < HW; i += 256) xpl[i] = xp[i];
  __syncthreads();
  const float TP = 6.28318530717958647692f;
  // G[kh,w] = sum_h x[h,w] e^{-i 2pi kh h/48}
  for (int t = tid; t < 6 * WD; t += 256) {
    int kh = t / WD, w = t % WD;
    float sr = 0.f, si = 0.f;
    for (int h = 0; h < WD; ++h) {
      float ang = -TP * (float)((kh * h) % WD) / (float)WD;
      float s, c; __sincosf(ang, &s, &c);
      float v = xpl[h * WD + w];
      sr += v * c; si += v * s;
    }
    Gr[t] = sr; Gi[t] = si;
  }
  __syncthreads();
  // Y[kh,kw] = sum_w G[kh,w] e^{-i 2pi kw w/48}
  if (tid < 36) {
    int kh = tid / 6, kw = tid % 6;
    float sr = 0.f, si = 0.f;
    for (int w = 0; w < WD; ++w) {
      float ang = -TP * (float)((kw * w) % WD) / (float)WD;
      float s, c; __sincosf(ang, &s, &c);
      float gr = Gr[kh * WD + w], gi = Gi[kh * WD + w];
      sr += gr * c - gi * s;
      si += gr * s + gi * c;
    }
    Yr[tid] = sr; Yi[tid] = si;
  }
  __syncthreads();
  // Q[kh,w] = Y[kh,0] + 2*sum_{kw=1..5} Y[kh,kw] e^{+i 2pi kw w/48}
  for (int t = tid; t < 6 * WD; t += 256) {
    int kh = t / WD, w = t % WD;
    float qr = Yr[kh * 6], qi = Yi[kh * 6];
    for (int kw = 1; kw < 6; ++kw) {
      float ang = TP * (float)((kw * w) % WD) / (float)WD;
      float s, c; __sincosf(ang, &s, &c);
      float yr = Yr[kh * 6 + kw], yi = Yi[kh * 6 + kw];
      qr += 2.f * (yr * c - yi * s);
      qi += 2.f * (yr * s + yi * c);
    }
    Qr[t] = qr; Qi[t] = qi;
  }
  __syncthreads();
  // low[h,w] = (1/2304) Re( sum_kh e^{+i 2pi kh h/48} Q[kh,w] )
  for (int pidx = tid; pidx < HW; pidx += 256) {
    int h = pidx / WD, w = pidx % WD;
    float acc = 0.f;
    for (int kh = 0; kh < 6; ++kh) {
      float ang = TP * (float)((kh * h) % WD) / (float)WD;
      float s, c; __sincosf(ang, &s, &c);
      acc += Qr[kh * WD + w] * c - Qi[kh * WD + w] * s;
    }
    low[(size_t)bc * HW + pidx] = acc * (1.f / (float)HW);
  }
}

// feat = dw3x3(low, lf) + dw3x3_dil2(x - low, hf)  (high = x - low exactly)
__global__ void freq_feat_k(const float* __restrict__ x, const float* __restrict__ low,
                            const float* __restrict__ lfw, const float* __restrict__ lfb,
                            const float* __restrict__ hfw, const float* __restrict__ hfb,
                            float* __restrict__ feat, int total)
{
  int idx = blockIdx.x * blockDim.x + threadIdx.x;
  if (idx >= total) return;
  int p = idx % HW;
  int c = (idx / HW) % 96;
  size_t plane = (size_t)(idx / HW) * HW;
  const float* pl = low + plane;
  const float* px = x + plane;
  int h = p / WD, w = p % WD;
  float s = lfb[c] + hfb[c];
  for (int kh = 0; kh < 3; ++kh) {
    int ih = h - 1 + kh;
    if (ih < 0 || ih >= WD) continue;
    for (int kw = 0; kw < 3; ++kw) {
      int iw = w - 1 + kw;
      if (iw < 0 || iw >= WD) continue;
      s += pl[ih * WD + iw] * lfw[c * 9 + kh * 3 + kw];
    }
  }
  for (int kh = 0; kh < 3; ++kh) {
    int ih = h - 2 + 2 * kh;
    if (ih < 0 || ih >= WD) continue;
    for (int kw = 0; kw < 3; ++kw) {
      int iw = w - 2 + 2 * kw;
      if (iw < 0 || iw >= WD) continue;
      s += (px[ih * WD + iw] - pl[ih * WD + iw]) * hfw[c * 9 + kh * 3 + kw];
    }
  }
  feat[idx] = s;
}

__global__ void mean_hw_k(const float* __restrict__ in, float* __restrict__ out)
{
  __shared__ float s[256];
  int bc = blockIdx.x, tid = threadIdx.x;
  const float* p = in + (size_t)bc * HW;
  float a = 0.f;
  for (int i = tid; i < HW; i += 256) a += p[i];
  s[tid] = a; __syncthreads();
  for (int st = 128; st > 0; st >>= 1) { if (tid < st) s[tid] += s[tid + st]; __syncthreads(); }
  if (tid == 0) out[bc] = s[0] * (1.f / (float)HW);
}

// g(96) -> relu fc 24 -> sigmoid fc 96
__global__ void fc_att_k(const float* __restrict__ g, const float* __restrict__ w1,
                         const float* __restrict__ b1, const float* __restrict__ w2,
                         const float* __restrict__ b2, float* __restrict__ att)
{
  __shared__ float hid[24];
  int b = blockIdx.x, tid = threadIdx.x;
  if (tid < 24) {
    float s = b1[tid];
    for (int c = 0; c < 96; ++c) s += g[b * 96 + c] * w1[tid * 96 + c];
    hid[tid] = fmaxf(s, 0.f);
  }
  __syncthreads();
  if (tid < 96) {
    float s = b2[tid];
    for (int j = 0; j < 24; ++j) s += hid[j] * w2[tid * 24 + j];
    att[b * 96 + tid] = sigf(s);
  }
}

// g(96) -> relu fc 24 -> fc 2 -> softmax
__global__ void cg_gate_k(const float* __restrict__ g, const float* __restrict__ w1,
                          const float* __restrict__ b1, const float* __restrict__ w2,
                          const float* __restrict__ b2, float* __restrict__ gout)
{
  __shared__ float hid[24];
  int b = blockIdx.x, tid = threadIdx.x;
  if (tid < 24) {
    float s = b1[tid];
    for (int c = 0; c < 96; ++c) s += g[b * 96 + c] * w1[tid * 96 + c];
    hid[tid] = fmaxf(s, 0.f);
  }
  __syncthreads();
  if (tid == 0) {
    float o0 = b2[0], o1 = b2[1];
    for (int j = 0; j < 24; ++j) { o0 += hid[j] * w2[j]; o1 += hid[j] * w2[24 + j]; }
    float mx = fmaxf(o0, o1);
    float e0 = __expf(o0 - mx), e1 = __expf(o1 - mx);
    float inv = 1.f / (e0 + e1);
    gout[b * 2 + 0] = e0 * inv;
    gout[b * 2 + 1] = e1 * inv;
  }
}

// Sobel edge magnitude
__global__ void edge_k(const float* __restrict__ x, float* __restrict__ edge, int total)
{
  int idx = blockIdx.x * blockDim.x + threadIdx.x;
  if (idx >= total) return;
  int p = idx % HW;
  size_t plane = (size_t)(idx / HW) * HW;
  const float* px = x + plane;
  int h = p / WD, w = p % WD;
  const float SX[9] = {-1.f, 0.f, 1.f, -2.f, 0.f, 2.f, -1.f, 0.f, 1.f};
  const float SY[9] = {-1.f, -2.f, -1.f, 0.f, 0.f, 0.f, 1.f, 2.f, 1.f};
  float ex = 0.f, ey = 0.f;
  for (int kh = 0; kh < 3; ++kh) {
    int ih = h - 1 + kh;
    if (ih < 0 || ih >= WD) continue;
    for (int kw = 0; kw < 3; ++kw) {
      int iw = w - 1 + kw;
      if (iw < 0 || iw >= WD) continue;
      float v = px[ih * WD + iw];
      ex += v * SX[kh * 3 + kw];
      ey += v * SY[kh * 3 + kw];
    }
  }
  edge[idx] = sqrtf(ex * ex + ey * ey + 1e-6f);
}

// ======================= selective scan =======================
// block = (b,k); thread = channel c; L sequential; x_dbl column staged in LDS;
// delta computed in-register; 4 directions scattered via global f32 atomics.
__global__ void scan_k(const float* __restrict__ u, const float* __restrict__ xdbl,
                       const float* __restrict__ dt_w, const float* __restrict__ dt_b,
                       const float* __restrict__ Alog, const float* __restrict__ Dp,
                       float* __restrict__ yout, int di, int R, int Cp)
{
  __shared__ float col[40];
  int b = blockIdx.x >> 2;
  int k = blockIdx.x & 3;
  int c = threadIdx.x;
  float A[16];
#pragma unroll
  for (int n = 0; n < 16; ++n) A[n] = -__expf(Alog[(size_t)(k * di + c) * 16 + n]);
  float dtw[8];
  for (int r = 0; r < R; ++r) dtw[r] = dt_w[(size_t)(k * di + c) * R + r];
  float dtb = dt_b[k * di + c];
  float Dv  = Dp[k * di + c];
  float hst[16];
#pragma unroll
  for (int n = 0; n < 16; ++n) hst[n] = 0.f;
  const float* xdb = xdbl + ((size_t)b * 4 + k) * Cp * HW;
  const float* up  = u + (size_t)b * di * HW + (size_t)c * HW;
  float*       yp  = yout + (size_t)b * di * HW + (size_t)c * HW;
  for (int l = 0; l < HW; ++l) {
    if (c < Cp) col[c] = xdb[(size_t)c * HW + l];
    __syncthreads();
    float t = dtb;
    for (int r = 0; r < R; ++r) t += dtw[r] * col[r];
    float delta = (t > 20.f) ? t : log1pf(__expf(t));
    int src = dir_idx(k, l);
    float uval = up[src];
    float du = delta * uval;
    float yv = 0.f;
#pragma unroll
    for (int n = 0; n < 16; ++n) {
      float hn = __expf(delta * A[n]) * hst[n] + du * col[R + n];
      hst[n] = hn;
      yv += hn * col[R + 16 + n];
    }
    atomicAdd(&yp[src], yv + Dv * uval);
    __syncthreads();
  }
}

// per-pixel layernorm over di channels, * ln_g + ln_b, * silu(z); in place on y
__global__ void lnorm_k(float* __restrict__ y, const float* __restrict__ xz,
                        const float* __restrict__ lng, const float* __restrict__ lnb, int di)
{
  __shared__ float s1[192], s2[192], mv[2];
  int bl = blockIdx.x;
  int b = bl / HW, l = bl % HW;
  int c = threadIdx.x;
  float v = y[(size_t)b * di * HW + (size_t)c * HW + l];
  s1[c] = v; s2[c] = v * v;
  __syncthreads();
  if (c == 0) {
    float a = 0.f, q = 0.f;
    for (int i = 0; i < di; ++i) { a += s1[i]; q += s2[i]; }
    float mean = a / (float)di;
    mv[0] = mean; mv[1] = q / (float)di - mean * mean;
  }
  __syncthreads();
  float mean = mv[0], var = mv[1];
  float z = xz[(size_t)b * 2 * di * HW + (size_t)(di + c) * HW + l];
  float o = (v - mean) * rsqrtf(var + 1e-5f) * lng[c] + lnb[c];
  o *= z * sigf(z);
  y[(size_t)b * di * HW + (size_t)c * HW + l] = o;
}

// ======================= elementwise helpers =======================
__global__ void zero_k(float* p, int n)
{ int i = blockIdx.x * blockDim.x + threadIdx.x; if (i < n) p[i] = 0.f; }

__global__ void mul2_k(const float* __restrict__ a, int aBstr, int aCoff,
                       const float* __restrict__ b, int bBstr, int bCoff,
                       float* __restrict__ o, int oBstr, int oCoff, int Cch, int total)
{
  int idx = blockIdx.x * blockDim.x + threadIdx.x;
  if (idx >= total) return;
  int p = idx % HW;
  int c = (idx / HW) % Cch;
  int bb = idx / (HW * Cch);
  o[(size_t)bb * oBstr + (size_t)(oCoff + c) * HW + p] =
      a[(size_t)bb * aBstr + (size_t)(aCoff + c) * HW + p] *
      b[(size_t)bb * bBstr + (size_t)(bCoff + c) * HW + p];
}

__global__ void scalefeat_k(const float* __restrict__ feat, const float* __restrict__ att,
                            float* __restrict__ cat, int total)
{
  int idx = blockIdx.x * blockDim.x + threadIdx.x;
  if (idx >= total) return;
  int p = idx % HW;
  int c = (idx / HW) % 96;
  int b = idx / (HW * 96);
  cat[(size_t)b * 192 * HW + (size_t)(96 + c) * HW + p] = feat[idx] * att[b * 96 + c];
}

__global__ void comb_k(const float* __restrict__ st1, const float* __restrict__ cross,
                       const float* __restrict__ g2, float* __restrict__ o, int total)
{
  int idx = blockIdx.x * blockDim.x + threadIdx.x;
  if (idx >= total) return;
  int b = idx / (HW * 96);
  o[idx] = g2[b * 2] * st1[idx] + g2[b * 2 + 1] * cross[idx];
}

__global__ void final_k(const float* __restrict__ ff, const float* __restrict__ ea,
                        float* __restrict__ out, int total)
{
  int idx = blockIdx.x * blockDim.x + threadIdx.x;
  if (idx >= total) return;
  out[idx] = ff[idx] * (1.f + ea[idx]);
}

// ======================= host orchestration =======================
static void gemm(const float* in, const float* w, const float* bias, float* out,
                 int Cin, int Cout, int inBstr, int inCoff, int outBstr, int outCoff,
                 int act, int dir, hipStream_t s)
{
  dim3 g(NB * HW / 32, cdiv(Cout, 16));
  gemm_wmma_k<<<g, dim3(32), 0, s>>>(in, w, bias, out, Cin, Cout, inBstr, inCoff,
                                     outBstr, outCoff, act, dir);
}

static void run_ss2d(const float* xin, int d, int di, int R, int Cp,
                     const float* in_w, const float* conv_w, const float* conv_b,
                     const float* xproj_w, const float* dt_w, const float* dt_b,
                     const float* Alog, const float* Dp, const float* ln_g,
                     const float* ln_b, const float* out_w,
                     float* xzb, float* ub, float* xdb, float* yb, float* dest,
                     hipStream_t s)
{
  // xz = x @ in_w.T    (B, 2di, HW)
  gemm(xin, in_w, nullptr, xzb, d, 2 * di, d * HW, 0, 2 * di * HW, 0, 0, 0, s);
  // u = silu(dwconv3x3(xz[:, :di]))
  int tot = NB * di * HW;
  dwconv_k<<<cdiv(tot, 256), 256, 0, s>>>(xzb, conv_w, conv_b, ub, di,
                                          2 * di * HW, 0, di * HW, 0, 3, 1, 1, 1, tot);
  // x_dbl[k] = xs[k] @ xproj_w[k].T  via direction-indexed reads of u
  for (int k = 0; k < 4; ++k)
    gemm(ub, xproj_w + (size_t)k * Cp * di, nullptr, xdb, di, Cp,
         di * HW, 0, 4 * Cp * HW, k * Cp, 0, k, s);
  // scan (all 4 directions scatter-add into yb)
  zero_k<<<cdiv(tot, 256), 256, 0, s>>>(yb, tot);
  scan_k<<<NB * 4, di, 0, s>>>(ub, xdb, dt_w, dt_b, Alog, Dp, yb, di, R, Cp);
  // layernorm * silu(z), in place
  lnorm_k<<<NB * HW, di, 0, s>>>(yb, xzb, ln_g, ln_b, di);
  // out = y @ out_w.T
  gemm(yb, out_w, nullptr, dest, di, d, di * HW, 0, d * HW, 0, 0, 0, s);
}

extern "C" void kernel_launch(void* const* d_in, const int* in_sizes, int n_in,
                              void* d_out, int out_size, void* d_ws, size_t ws_size,
                              hipStream_t stream)
{
  (void)in_sizes; (void)n_in; (void)out_size; (void)ws_size;
  const float* x         = (const float*)d_in[0];
  const float* proj_in_w = (const float*)d_in[1];
  const float* proj_in_b = (const float*)d_in[2];
  const float* dw_w      = (const float*)d_in[3];
  const float* dw_b      = (const float*)d_in[4];
  const float* pw0_w     = (const float*)d_in[5];
  const float* pw0_b     = (const float*)d_in[6];
  const float* proj_out_w= (const float*)d_in[7];
  const float* proj_out_b= (const float*)d_in[8];
  const float* fa1_w     = (const float*)d_in[9];
  const float* fa1_b     = (const float*)d_in[10];
  const float* fa2_w     = (const float*)d_in[11];
  const float* fa2_b     = (const float*)d_in[12];
  const float* lf_w      = (const float*)d_in[13];
  const float* lf_b      = (const float*)d_in[14];
  const float* hf_w      = (const float*)d_in[15];
  const float* hf_b      = (const float*)d_in[16];
  const float* ff_w      = (const float*)d_in[17];
  const float* ff_b      = (const float*)d_in[18];
  const float* ea1_w     = (const float*)d_in[19];
  const float* ea1_b     = (const float*)d_in[20];
  const float* ea2_w     = (const float*)d_in[21];
  const float* ea2_b     = (const float*)d_in[22];
  const float* cg11_w    = (const float*)d_in[23];
  const float* cg11_b    = (const float*)d_in[24];
  const float* cg12_w    = (const float*)d_in[25];
  const float* cg12_b    = (const float*)d_in[26];
  const float* sp0_w     = (const float*)d_in[27];
  const float* sp0_b     = (const float*)d_in[28];
  // s0 (d=48, di=96, R=3, Cp=35)
  const float* s0_in_w   = (const float*)d_in[29];
  const float* s0_conv_w = (const float*)d_in[30];
  const float* s0_conv_b = (const float*)d_in[31];
  const float* s0_xproj  = (const float*)d_in[32];
  const float* s0_dt_w   = (const float*)d_in[33];
  const float* s0_dt_b   = (const float*)d_in[34];
  const float* s0_Alog   = (const float*)d_in[35];
  const float* s0_D      = (const float*)d_in[36];
  const float* s0_ln_g   = (const float*)d_in[37];
  const float* s0_ln_b   = (const float*)d_in[38];
  const float* s0_out_w  = (const float*)d_in[39];
  // s1 (d=96, di=192, R=6, Cp=38)
  const float* s1_in_w   = (const float*)d_in[40];
  const float* s1_conv_w = (const float*)d_in[41];
  const float* s1_conv_b = (const float*)d_in[42];
  const float* s1_xproj  = (const float*)d_in[43];
  const float* s1_dt_w   = (const float*)d_in[44];
  const float* s1_dt_b   = (const float*)d_in[45];
  const float* s1_Alog   = (const float*)d_in[46];
  const float* s1_D      = (const float*)d_in[47];
  const float* s1_ln_g   = (const float*)d_in[48];
  const float* s1_ln_b   = (const float*)d_in[49];
  const float* s1_out_w  = (const float*)d_in[50];

  // ---- workspace layout (floats) ----
  float* ws = (float*)d_ws;
  size_t off = 0;
  auto alloc = [&](size_t n) { float* q = ws + off; off += n; return q; };
  const size_t BHW = (size_t)NB * HW;
  float* low    = alloc(BHW * 96);
  float* feat   = alloc(BHW * 96);
  float* edgeb  = alloc(BHW * 96);
  float* midb   = alloc(BHW * 24);
  float* eatt   = alloc(BHW * 96);
  float* fused  = alloc(BHW * 192);
  float* dwb    = alloc(BHW * 144);
  float* hbuf   = alloc(BHW * 96);
  float* st0    = alloc(BHW * 48);
  float* st1    = alloc(BHW * 96);
  float* crossb = alloc(BHW * 96);
  float* catb   = alloc(BHW * 192);
  float* gvec   = alloc(NB * 192);
  float* avec   = alloc(NB * 96);
  float* g2     = alloc(16);
  float* xzb    = alloc(BHW * 384);
  float* ubuf   = alloc(BHW * 192);
  float* xdb    = alloc(BHW * 4 * 38);
  float* ybuf   = alloc(BHW * 192);

  const int T96  = NB * 96 * HW;
  const int T48  = NB * 48 * HW;
  const int T144 = NB * 144 * HW;

  // ---- frequency branch ----
  freq_low_k<<<NB * 96, 256, 0, stream>>>(x, low);
  freq_feat_k<<<cdiv(T96, 256), 256, 0, stream>>>(x, low, lf_w, lf_b, hf_w, hf_b, feat, T96);
  mean_hw_k<<<NB * 96, 256, 0, stream>>>(feat, gvec);
  fc_att_k<<<NB, 96, 0, stream>>>(gvec, fa1_w, fa1_b, fa2_w, fa2_b, avec);
  scalefeat_k<<<cdiv(T96, 256), 256, 0, stream>>>(feat, avec, catb, T96);

  // ---- edge branch ----
  edge_k<<<cdiv(T96, 256), 256, 0, stream>>>(x, edgeb, T96);
  gemm(edgeb, ea1_w, ea1_b, midb, 96, 24, 96 * HW, 0, 24 * HW, 0, 1, 0, stream);
  gemm(midb, ea2_w, ea2_b, eatt, 24, 96, 24 * HW, 0, 96 * HW, 0, 2, 0, stream);

  // ---- main path ----
  gemm(x, proj_in_w, proj_in_b, fused, 96, 192, 96 * HW, 0, 192 * HW, 0, 0, 0, stream);
  dwconv_k<<<cdiv(T144, 256), 256, 0, stream>>>(fused, dw_w, dw_b, dwb, 144,
                                                192 * HW, 48, 144 * HW, 0, 7, 3, 1, 0, T144);
  // h = pwa * dw0
  mul2_k<<<cdiv(T48, 256), 256, 0, stream>>>(fused, 192 * HW, 0, dwb, 144 * HW, 0,
                                             hbuf, 48 * HW, 0, 48, T48);
  // ss2d #0
  run_ss2d(hbuf, 48, 96, 3, 35, s0_in_w, s0_conv_w, s0_conv_b, s0_xproj, s0_dt_w,
           s0_dt_b, s0_Alog, s0_D, s0_ln_g, s0_ln_b, s0_out_w,
           xzb, ubuf, xdb, ybuf, st0, stream);
  // h2 = (pw0(st0)+b) * dw1
  gemm(st0, pw0_w, pw0_b, fused, 48, 96, 48 * HW, 0, 96 * HW, 0, 0, 0, stream);
  mul2_k<<<cdiv(T96, 256), 256, 0, stream>>>(fused, 96 * HW, 0, dwb, 144 * HW, 48,
                                             hbuf, 96 * HW, 0, 96, T96);
  // ss2d #1
  run_ss2d(hbuf, 96, 192, 6, 38, s1_in_w, s1_conv_w, s1_conv_b, s1_xproj, s1_dt_w,
           s1_dt_b, s1_Alog, s1_D, s1_ln_g, s1_ln_b, s1_out_w,
           xzb, ubuf, xdb, ybuf, st1, stream);
  // channel gate
  mean_hw_k<<<NB * 96, 256, 0, stream>>>(st1, gvec);
  cg_gate_k<<<NB, 32, 0, stream>>>(gvec, cg11_w, cg11_b, cg12_w, cg12_b, g2);
  // cross = sp0(st0)
  gemm(st0, sp0_w, sp0_b, crossb, 48, 96, 48 * HW, 0, 96 * HW, 0, 0, 0, stream);
  // out = g0*st1 + g1*cross  -> ubuf (free after ss2d)
  comb_k<<<cdiv(T96, 256), 256, 0, stream>>>(st1, crossb, g2, ubuf, T96);
  // proj_out into cat[:, :96]; freq already in cat[:, 96:]
  gemm(ubuf, proj_out_w, proj_out_b, catb, 96, 96, 96 * HW, 0, 192 * HW, 0, 0, 0, stream);
  // ff over concat
  gemm(catb, ff_w, ff_b, ybuf, 192, 96, 192 * HW, 0, 96 * HW, 0, 0, 0, stream);
  // out = ff * edge_att + ff
  final_k<<<cdiv(T96, 256), 256, 0, stream>>>(ybuf, eatt, (float*)d_out, T96);
}